// RGBDerivativeConv_420906795455
// MI455X (gfx1250) — compile-verified
//
#include <hip/hip_runtime.h>
#include <stdint.h>

// Depthwise 3x3 derivative conv, fp32, (32,3,512,512), pad 1.
// Separable: out(y,x) = colsum(y,x+1) - colsum(y,x-1),
//            colsum(y,x) = in(y-1,x)+in(y,x)+in(y+1,x).
// Memory-bound (201 MB @ 23.3 TB/s ~ 8.7 us floor). Strategy:
//  - async global->LDS b128 double-buffered row staging (ASYNCcnt)
//  - vertical ones-band sum via V_WMMA_F32_16X16X4_F32 (T[16x18] x X[18x16])
//  - horizontal diff = 1 VALU sub, b128 stores.

#define W       512
#define H       512
#define PLANES  96          // 32 batches * 3 channels
#define RBLK    8           // row-blocks per plane (64 rows each)
#define TILES   4           // 16-row tiles per block
#define TROWS   16
#define INROWS  18          // 16 + 2 halo rows
#define LDSW    520         // row stride (floats), bank-padded
#define CSW     516         // colsum row stride (floats)
#define INBUF_FLOATS (INROWS * LDSW)          // 9360
#define CS_OFF       (2 * INBUF_FLOATS)       // 18720
#define SMEM_FLOATS  (CS_OFF + TROWS * CSW)   // 26976
#define SMEM_BYTES   (SMEM_FLOATS * 4)        // 107904 B  (<320KB WGP LDS)

typedef __attribute__((ext_vector_type(2))) float v2f;
typedef __attribute__((ext_vector_type(4))) float f4;
typedef __attribute__((ext_vector_type(8))) float v8f;
typedef int v4i __attribute__((ext_vector_type(4)));

typedef __attribute__((address_space(1))) v4i* gp_v4i;
typedef __attribute__((address_space(3))) v4i* lp_v4i;

__device__ __forceinline__ void async_b128(const float* gsrc, float* ldst) {
#if defined(__gfx1250__) && __has_builtin(__builtin_amdgcn_global_load_async_to_lds_b128)
    __builtin_amdgcn_global_load_async_to_lds_b128(
        (gp_v4i)gsrc, (lp_v4i)ldst, 0, 0);
#else
    *(f4*)ldst = *(const f4*)gsrc;       // synchronous fallback
#endif
}

__device__ __forceinline__ void wait_async() {
#if defined(__gfx1250__) && __has_builtin(__builtin_amdgcn_s_wait_asynccnt)
    __builtin_amdgcn_s_wait_asynccnt(0);
#endif
}

__global__ void __launch_bounds__(256)
RGBDerivativeConv_kernel(const float* __restrict__ x, float* __restrict__ out) {
    extern __shared__ float smem[];

    const int tid   = threadIdx.x;
    const int plane = blockIdx.x / RBLK;          // b*3 + c
    const int rblk  = blockIdx.x % RBLK;
    const int row0  = rblk * (TILES * TROWS);     // first output row of block

    const float* gin  = x   + (size_t)plane * (size_t)(W * H);
    float*       gout = out + (size_t)plane * (size_t)(W * H);

    const int lane = tid & 31;
    const int wv   = tid >> 5;      // wave id 0..7
    const int m    = lane & 15;     // A-row / B-col within half-wave
    const int hi   = lane >> 4;     // half-wave select

    // A fragment: T[16x18] ones band (T[i][i..i+2]=1), zero padded to K=20.
    // 32-bit A 16x4 layout: VGPR0 = {K=k0 lanes0-15, K=k0+2 lanes16-31}, VGPR1 = K+1.
    // A is zero for K>=18, so B may be loaded from a clamped row (no predication).
    v2f afrag[5];
#pragma unroll
    for (int s = 0; s < 5; ++s) {
        int j0 = 4 * s + 2 * hi;
        afrag[s].x = (j0     >= m && j0     <= m + 2 && j0     < 18) ? 1.f : 0.f;
        afrag[s].y = (j0 + 1 >= m && j0 + 1 <= m + 2 && j0 + 1 < 18) ? 1.f : 0.f;
    }

    auto load_tile = [&](int t, int buf) {
        float* dst = smem + buf * INBUF_FLOATS;
        int grow0 = row0 + t * TROWS - 1;                 // first halo row
        f4 z = {0.f, 0.f, 0.f, 0.f};
#pragma unroll
        for (int i = 0; i < 9; ++i) {                     // 18 rows * 128 f4 / 256 thr
            int idx = tid + 256 * i;
            int r   = idx >> 7;                           // 0..17 (wave-uniform)
            int q   = idx & 127;                          // f4 column
            int gr  = grow0 + r;
            float* l = dst + r * LDSW + q * 4;
            if (gr >= 0 && gr < H) {
                async_b128(gin + (size_t)gr * W + q * 4, l);
            } else {
                *(f4*)l = z;                              // zero-pad halo row
            }
        }
    };

    load_tile(0, 0);
    wait_async();
    __syncthreads();

    for (int t = 0; t < TILES; ++t) {
        const int p = t & 1;
        if (t + 1 < TILES) load_tile(t + 1, 1 - p);       // overlap next tile's DMA

        const float* inb = smem + p * INBUF_FLOATS;
        float*       cs  = smem + CS_OFF;

        // ---- WMMA stage: colsum = T x Xrows, 32 column-chunks, 4 per wave ----
#pragma unroll
        for (int j = 0; j < 4; ++j) {
            const int cx = (wv * 4 + j) * 16;
            v8f acc = {0.f, 0.f, 0.f, 0.f, 0.f, 0.f, 0.f, 0.f};
#pragma unroll
            for (int s = 0; s < 5; ++s) {
                int k0 = 4 * s + 2 * hi;
                int r0 = k0     < 17 ? k0     : 17;   // clamp; A==0 for K>=18
                int r1 = k0 + 1 < 17 ? k0 + 1 : 17;
                v2f b;
                b.x = inb[r0 * LDSW + cx + m];
                b.y = inb[r1 * LDSW + cx + m];
                acc = __builtin_amdgcn_wmma_f32_16x16x4_f32(
                          false, afrag[s], false, b, (short)0, acc, false, false);
            }
            // D layout: VGPR v -> row (v + 8*hi), col = cx + m. cs col offset +1.
#pragma unroll
            for (int v = 0; v < 8; ++v) {
                cs[(v + 8 * hi) * CSW + (cx + m + 1)] = acc[v];
            }
        }
        if (tid < TROWS) {                 // colsum(-1)=colsum(512)=0 (zero pad)
            cs[tid * CSW + 0]   = 0.f;
            cs[tid * CSW + 513] = 0.f;
            cs[tid * CSW + 514] = 0.f;
            cs[tid * CSW + 515] = 0.f;
        }
        __syncthreads();

        // ---- horizontal diff + b128 store: out(y,x) = cs[x+2] - cs[x] ----
#pragma unroll
        for (int i = 0; i < 8; ++i) {
            int q  = tid + 256 * i;        // 16 rows * 128 f4
            int y  = q >> 7;
            int x4 = (q & 127) * 4;
            const float* row = cs + y * CSW;
            f4 L = *(const f4*)(row + x4);
            f4 R = *(const f4*)(row + x4 + 4);
            f4 o;
            o.x = L.z - L.x;
            o.y = L.w - L.y;
            o.z = R.x - L.z;
            o.w = R.y - L.w;
            *(f4*)(gout + (size_t)(row0 + t * TROWS + y) * W + x4) = o;
        }

        if (t + 1 < TILES) wait_async();
        __syncthreads();
    }
}

extern "C" void kernel_launch(void* const* d_in, const int* in_sizes, int n_in,
                              void* d_out, int out_size, void* d_ws, size_t ws_size,
                              hipStream_t stream) {
    (void)in_sizes; (void)n_in; (void)out_size; (void)d_ws; (void)ws_size;
    const float* x = (const float*)d_in[0];
    // d_in[1] (filters) is the fixed separable [-1,0,1] x [1,1,1] kernel; baked in.
    float* out = (float*)d_out;
    dim3 grid(PLANES * RBLK);   // 768 blocks
    dim3 block(256);            // 8 wave32
    RGBDerivativeConv_kernel<<<grid, block, SMEM_BYTES, stream>>>(x, out);
}